// RuntimeGeometryCalculation_25314537242666
// MI455X (gfx1250) — compile-verified
//
#include <hip/hip_runtime.h>

typedef __attribute__((ext_vector_type(2))) float v2f;
typedef __attribute__((ext_vector_type(8))) float v8f;

// ---------------------------------------------------------------------------
// Kernel 1: zero the direction_units accumulator (output region 3).
// ---------------------------------------------------------------------------
__global__ void geo_zero_f32(float* __restrict__ p, int n) {
  int i = blockIdx.x * blockDim.x + threadIdx.x;
  if (i < n) p[i] = 0.0f;
}

// ---------------------------------------------------------------------------
// Kernel 2: per-edge unit vectors, scatter-add +/- into direction_units.
// Accumulator is N*3*4B = 1.2MB -> L2-resident, f32 atomics are cheap.
// ---------------------------------------------------------------------------
__global__ void geo_edge_scatter(const float* __restrict__ pos,
                                 const int* __restrict__ ei,
                                 float* __restrict__ du,
                                 int E) {
  int e = blockIdx.x * blockDim.x + threadIdx.x;
  if (e >= E) return;
  int r = ei[e];
  int c = ei[(long long)E + e];
  float ex = pos[3 * c + 0] - pos[3 * r + 0];
  float ey = pos[3 * c + 1] - pos[3 * r + 1];
  float ez = pos[3 * c + 2] - pos[3 * r + 2];
  float dist = sqrtf((ex * ex + ey * ey) + ez * ez) + 1e-8f;
  float ux = ex / dist, uy = ey / dist, uz = ez / dist;
  atomicAdd(&du[3 * r + 0], ux);
  atomicAdd(&du[3 * r + 1], uy);
  atomicAdd(&du[3 * r + 2], uz);
  atomicAdd(&du[3 * c + 0], -ux);
  atomicAdd(&du[3 * c + 1], -uy);
  atomicAdd(&du[3 * c + 2], -uz);
}

// ---------------------------------------------------------------------------
// Kernel 3: angular_info[n,h] = |du[n]|^2 broadcast over H.
// float4 stores: 32 consecutive lanes write 512B contiguous.
// ---------------------------------------------------------------------------
__global__ void geo_angular_fill(const float* __restrict__ du,
                                 float* __restrict__ ang,
                                 int N, int H, int hq) {
  int t = blockIdx.x * blockDim.x + threadIdx.x;
  int n = t / hq;
  int q = t - n * hq;
  if (n >= N) return;
  float x = du[3 * n + 0], y = du[3 * n + 1], z = du[3 * n + 2];
  float s = (x * x + y * y) + z * z;   // reference summation order
  float4 v; v.x = s; v.y = s; v.z = s; v.w = s;
  *(float4*)(ang + (long long)n * H + (long long)q * 4) = v;
}

// ---------------------------------------------------------------------------
// Per-edge dihedral scalar (shared helper; bit-identical op order everywhere).
// ---------------------------------------------------------------------------
__device__ __forceinline__ float geo_dihedral_scalar(const float* __restrict__ pos,
                                                     const int* __restrict__ ei,
                                                     const float* __restrict__ du,
                                                     int E, long long e) {
  int r = ei[e];
  int c = ei[(long long)E + e];
  float ex = pos[3 * c + 0] - pos[3 * r + 0];
  float ey = pos[3 * c + 1] - pos[3 * r + 1];
  float ez = pos[3 * c + 2] - pos[3 * r + 2];
  float dist = sqrtf((ex * ex + ey * ey) + ez * ez) + 1e-8f;
  float ux = ex / dist, uy = ey / dist, uz = ez / dist;
  float vix = du[3 * r + 0], viy = du[3 * r + 1], viz = du[3 * r + 2];
  float vjx = du[3 * c + 0], vjy = du[3 * c + 1], vjz = du[3 * c + 2];
  // Match reference ops/order: dot_vi = vi.u ; dot_vj = vj.(-u)
  float dvi = (vix * ux + viy * uy) + viz * uz;
  float nux = -ux, nuy = -uy, nuz = -uz;
  float dvj = (vjx * nux + vjy * nuy) + vjz * nuz;
  float w1x = vix - dvi * ux,  w1y = viy - dvi * uy,  w1z = viz - dvi * uz;
  float w2x = vjx - dvj * nux, w2y = vjy - dvj * nuy, w2z = vjz - dvj * nuz;
  return (w1x * w2x + w1y * w2y) + w1z * w2z;
}

// ---------------------------------------------------------------------------
// Kernel 4 (HH compile-time): dihedral broadcast along H via WMMA rank-1
// product. One wave32 per 16 edges (full tiles only; tail handled separately).
// A(16x4) = dihedral scalar replicated in K, B(4x16) = one-hot K row of ones
// -> D[m,n] = d_m * 1.0 bit-exact. D layout: VGPR v holds row M=v+8*(lane>=16),
// column N = lane%16. All store offsets are compile-time immediates.
// ---------------------------------------------------------------------------
template <int HH>
__global__ void geo_dihedral_wmma(const float* __restrict__ pos,
                                  const int* __restrict__ ei,
                                  const float* __restrict__ du,
                                  float* __restrict__ dih,
                                  int E, int nTiles) {
  int gtid = blockIdx.x * blockDim.x + threadIdx.x;
  int wave = gtid >> 5;
  int lane = threadIdx.x & 31;
  if (wave >= nTiles) return;          // wave-uniform: EXEC stays all-1s below
  long long e0 = (long long)wave * 16;

  int m  = lane & 15;                  // edge within tile (lanes 16-31 mirror 0-15)
  int hi = lane >> 4;
  float d = geo_dihedral_scalar(pos, ei, du, E, e0 + m);

  // Rank-1 broadcast on the matrix pipe: D[m,n] = d_m * 1.0 (exact).
  v2f A; A.x = d; A.y = d;                          // replicate across all K
  v2f B; B.x = (hi == 0) ? 1.0f : 0.0f; B.y = 0.0f; // one-hot K row of ones
  v8f C = {};
  v8f D = __builtin_amdgcn_wmma_f32_16x16x4_f32(
      /*neg_a=*/false, A, /*neg_b=*/false, B,
      /*c_mod=*/(short)0, C, /*reuse_a=*/false, /*reuse_b=*/false);

  // Single 64-bit base per lane; 64 stores with immediate offsets.
  float* p = dih + ((long long)e0 + (hi << 3)) * HH + m;
#pragma unroll
  for (int v = 0; v < 8; ++v) {
#pragma unroll
    for (int h = 0; h < HH; h += 16) {
      p[v * HH + h] = D[v];
    }
  }
}

// ---------------------------------------------------------------------------
// Generic dihedral fill (no WMMA): used for the E%16 tail and for H != 128.
// One thread per (edge, 4-column group), float4 stores.
// ---------------------------------------------------------------------------
__global__ void geo_dihedral_generic(const float* __restrict__ pos,
                                     const int* __restrict__ ei,
                                     const float* __restrict__ du,
                                     float* __restrict__ dih,
                                     int E, int H, int hq,
                                     int eStart, int eCount) {
  int t = blockIdx.x * blockDim.x + threadIdx.x;
  int ee = t / hq;
  int q  = t - ee * hq;
  if (ee >= eCount) return;
  long long e = (long long)eStart + ee;
  float d = geo_dihedral_scalar(pos, ei, du, E, e);
  float4 v; v.x = d; v.y = d; v.z = d; v.w = d;
  *(float4*)(dih + e * (long long)H + (long long)q * 4) = v;
}

// ---------------------------------------------------------------------------
// Host launcher. Outputs packed in d_out:
//   [0, N*H)                 angular_info
//   [N*H, N*H + E*H)         dihedral_info
//   [N*H + E*H, ... + N*3)   direction_units (also the accumulator)
// ---------------------------------------------------------------------------
extern "C" void kernel_launch(void* const* d_in, const int* in_sizes, int n_in,
                              void* d_out, int out_size, void* d_ws, size_t ws_size,
                              hipStream_t stream) {
  const float* pos = (const float*)d_in[0];
  const int*   ei  = (const int*)d_in[1];
  // d_in[2] (vector_features) is unused by the reference math.

  const int N = in_sizes[0] / 3;
  const int E = in_sizes[1] / 2;
  const int H = in_sizes[2] / in_sizes[0];   // (N*3*H)/(N*3)

  float* out = (float*)d_out;
  float* ang = out;
  float* dih = out + (long long)N * H;
  float* du  = out + (long long)N * H + (long long)E * H;

  // 1) zero accumulator (every call: harness replays without re-initializing)
  {
    int n = N * 3;
    geo_zero_f32<<<(n + 255) / 256, 256, 0, stream>>>(du, n);
  }
  // 2) scatter-add unit vectors
  geo_edge_scatter<<<(E + 255) / 256, 256, 0, stream>>>(pos, ei, du, E);
  // 3) angular broadcast (float4 stores)
  {
    int hq = H / 4;
    long long t = (long long)N * hq;
    geo_angular_fill<<<(int)((t + 255) / 256), 256, 0, stream>>>(du, ang, N, H, hq);
  }
  // 4) dihedral broadcast
  if (H == 128) {
    int nTiles = E / 16;                       // full 16-edge tiles (WMMA path)
    if (nTiles > 0) {
      long long thr = (long long)nTiles * 32;
      geo_dihedral_wmma<128><<<(int)((thr + 255) / 256), 256, 0, stream>>>(
          pos, ei, du, dih, E, nTiles);
    }
    int tail = E - nTiles * 16;                // E%16 leftover edges (0 here)
    if (tail > 0) {
      int hq = H / 4;
      long long t = (long long)tail * hq;
      geo_dihedral_generic<<<(int)((t + 255) / 256), 256, 0, stream>>>(
          pos, ei, du, dih, E, H, hq, nTiles * 16, tail);
    }
  } else {
    int hq = H / 4;
    long long t = (long long)E * hq;
    geo_dihedral_generic<<<(int)((t + 255) / 256), 256, 0, stream>>>(
        pos, ei, du, dih, E, H, hq, 0, E);
  }
}